// ClsContrastLoss_3908420239917
// MI455X (gfx1250) — compile-verified
//
#include <hip/hip_runtime.h>
#include <math.h>

#define BS      1024
#define DIM     1024
#define NEG     32
#define TEMP    0.05f
#define CLS_WT  0.2f
#define EPSF    1e-8f

typedef __attribute__((ext_vector_type(2))) float v2f;
typedef __attribute__((ext_vector_type(8))) float v8f;

// One block (2 waves, 64 threads) per batch row.
// Wave w computes q . neg_n and |neg_n|^2 for negs [16w, 16w+16) via
// V_WMMA_F32_16X16X4_F32 with A = q broadcast over M, B = 16 negs over N.
__global__ __launch_bounds__(64) void cls_row_kernel(
    const float* __restrict__ q,
    const float* __restrict__ p,
    const float* __restrict__ neg,
    float* __restrict__ row_loss)
{
    __shared__ float lds_q[DIM];
    __shared__ float red_qq[64], red_pp[64], red_qp[64];
    __shared__ float s_ann[64];
    __shared__ float s_y[NEG], s_nn[NEG];
    __shared__ float s_qq, s_pp, s_qp;

    const int b    = blockIdx.x;
    const int tid  = threadIdx.x;
    const int wave = tid >> 5;
    const int lane = tid & 31;

    const float* qrow = q + (size_t)b * DIM;
    const float* prow = p + (size_t)b * DIM;

    // ---- Stage A: stream q,p once; stage q into LDS; partial qq/pp/qp ----
    float aqq = 0.f, app = 0.f, aqp = 0.f;
    for (int i = tid; i < DIM / 4; i += 64) {
        const float4 qv = ((const float4*)qrow)[i];
        const float4 pv = ((const float4*)prow)[i];
        ((float4*)lds_q)[i] = qv;
        aqq += qv.x * qv.x + qv.y * qv.y + qv.z * qv.z + qv.w * qv.w;
        app += pv.x * pv.x + pv.y * pv.y + pv.z * pv.z + pv.w * pv.w;
        aqp += qv.x * pv.x + qv.y * pv.y + qv.z * pv.z + qv.w * pv.w;
    }
    red_qq[tid] = aqq; red_pp[tid] = app; red_qp[tid] = aqp;
    __syncthreads();
    if (tid == 0) {
        float sqq = 0.f, spp = 0.f, sqp = 0.f;
        for (int i = 0; i < 64; ++i) { sqq += red_qq[i]; spp += red_pp[i]; sqp += red_qp[i]; }
        s_qq = sqq; s_pp = spp; s_qp = sqp;
    }

    // ---- Stage B: WMMA over the 16 negs owned by this wave ----
    // A (16x4 f32): lanes 0-15 hold K=0,1; lanes 16-31 hold K=2,3  -> klo
    // B (4x16 f32): lane n / n+16 hold neg_n at K=0,1 / K=2,3      -> same klo
    const int nidx = lane & 15;
    const int klo  = (lane >> 4) * 2;
    const float* nrow = neg + ((size_t)b * NEG + (size_t)(wave * 16 + nidx)) * DIM + klo;
    const float* qk   = lds_q + klo;

    v8f acc = {0.f, 0.f, 0.f, 0.f, 0.f, 0.f, 0.f, 0.f};
    float ann = 0.f;
    #pragma unroll 4
    for (int d0 = 0; d0 < DIM; d0 += 4) {
        const v2f bv = *(const v2f*)(nrow + d0);   // neg_n[d0+klo .. +1]
        v2f av;
        av[0] = qk[d0];                             // q[d0+klo]
        av[1] = qk[d0 + 1];                         // q[d0+klo+1]
        acc = __builtin_amdgcn_wmma_f32_16x16x4_f32(
                  false, av, false, bv, (short)0, acc, false, false);
        ann += bv[0] * bv[0] + bv[1] * bv[1];       // |neg|^2 fused in same pass
    }
    s_ann[tid] = ann;
    __syncthreads();
    if (lane < 16) {
        const int ng = wave * 16 + lane;
        s_y[ng]  = acc[0];  // D[0, lane] == q . neg_ng (all D rows identical)
        s_nn[ng] = s_ann[wave * 32 + lane] + s_ann[wave * 32 + lane + 16];
    }
    __syncthreads();

    // ---- Stage C: 33-way log-softmax for this row ----
    if (tid == 0) {
        const float qn = fmaxf(sqrtf(s_qq), EPSF);
        const float pn = fmaxf(sqrtf(s_pp), EPSF);
        const float inv_t = 1.0f / TEMP;
        const float s0 = (s_qp / (qn * pn)) * inv_t;

        float m = s0;
        float sn[NEG];
        #pragma unroll
        for (int i = 0; i < NEG; ++i) {
            const float nnrm = fmaxf(sqrtf(s_nn[i]), EPSF);
            sn[i] = (s_y[i] / (qn * nnrm)) * inv_t;
            m = fmaxf(m, sn[i]);
        }
        float se = expf(s0 - m);
        #pragma unroll
        for (int i = 0; i < NEG; ++i) se += expf(sn[i] - m);
        // -log_softmax[0] = m + log(sum exp) - s0
        row_loss[b] = m + logf(se) - s0;
    }
}

// Deterministic final reduction of 1024 per-row losses.
__global__ __launch_bounds__(256) void cls_reduce_kernel(
    const float* __restrict__ row_loss, float* __restrict__ out)
{
    __shared__ float red[256];
    const int t = threadIdx.x;
    float a = row_loss[t] + row_loss[t + 256] + row_loss[t + 512] + row_loss[t + 768];
    red[t] = a;
    __syncthreads();
    for (int s = 128; s > 0; s >>= 1) {
        if (t < s) red[t] += red[t + s];
        __syncthreads();
    }
    if (t == 0) out[0] = red[0] * (CLS_WT / (float)BS);
}

extern "C" void kernel_launch(void* const* d_in, const int* in_sizes, int n_in,
                              void* d_out, int out_size, void* d_ws, size_t ws_size,
                              hipStream_t stream) {
    const float* q   = (const float*)d_in[0];  // [BS, D]
    const float* p   = (const float*)d_in[1];  // [BS, D]
    const float* neg = (const float*)d_in[2];  // [BS*NEG, D]
    float* row_loss  = (float*)d_ws;           // BS floats of scratch
    float* out       = (float*)d_out;          // 1 float

    cls_row_kernel<<<BS, 64, 0, stream>>>(q, p, neg, row_loss);
    cls_reduce_kernel<<<1, 256, 0, stream>>>(row_loss, out);
}